// DCT_15144054685932
// MI455X (gfx1250) — compile-verified
//
#include <hip/hip_runtime.h>
#include <hip/hip_bf16.h>

typedef __attribute__((ext_vector_type(2))) float v2f;
typedef __attribute__((ext_vector_type(8))) float v8f;

// ---------------------------------------------------------------------------
// Kernel 1: block DCT as GEMM  coeff[16384,64] = A[16384,64] * W[64,64]
// via V_WMMA_F32_16X16X4_F32 (full fp32 matrix pipe, K chained in steps of 4).
// One wave = one 16-row M-tile x full N=64 (4 accumulator tiles).
//
// The 64x64 basis is staged into LDS pre-swizzled so every lane's B fragment
// {W[r,c], W[r+1,c]} is a contiguous float2 -> one ds_load_b64 straight into
// the even-aligned VGPR pair the WMMA operand needs (no fixup movs).
//   float2 index: (t*8 + h*4 + n)*16 + col,  r = 4t + 2h,  c = 16n + col
// ---------------------------------------------------------------------------
__global__ void __launch_bounds__(256) dct_wmma_kernel(
    const float* __restrict__ in,     // [16,256,256,1]
    const float* __restrict__ basis,  // [8,8,1,64] flat = W[p,k], p=8x+y
    float* __restrict__ coeff)        // [16384,64]
{
    __shared__ float sB[64 * 64];     // 16 KB basis, swizzled into B-frag order

    const int tid = threadIdx.x;
    #pragma unroll
    for (int i = 0; i < 16; ++i) {    // 256 thr * 16 = 4096 elements
        const int src = tid + i * 256;
        const int r = src >> 6, c = src & 63;
        const int t = r >> 2, h = (r >> 1) & 1, par = r & 1;
        const int n = c >> 4, col = c & 15;
        sB[(((t * 8 + h * 4 + n) * 16 + col) << 1) + par] = basis[src];
    }
    __syncthreads();

    const int wave  = tid >> 5;                  // 0..7
    const int lane  = tid & 31;
    const int mtile = blockIdx.x * 8 + wave;     // 0..1023  (16384/16 tiles)

    // A-fragment row for this lane (16x4 f32 layout: both half-waves M=0..15)
    const int m  = mtile * 16 + (lane & 15);     // global block index
    const int b  = m >> 10;                      // batch
    const int hc = (m >> 5) & 31;                // block row
    const int wc = m & 31;                       // block col
    const int khalf = (lane >> 4) << 1;          // K offset: 0 or 2
    const int col   = lane & 15;                 // B/N column within tile

    // Per-lane base into swizzled basis (float2 units): h*64 + col
    const v2f* bp = ((const v2f*)sB) + ((lane >> 4) << 6) + col;
    const float* arow =
        in + ((size_t)(b * 256 + hc * 8) * 256 + wc * 8);  // block top-left

    v8f acc0 = {}, acc1 = {}, acc2 = {}, acc3 = {};

    #pragma unroll
    for (int t = 0; t < 16; ++t) {               // K = 64 in steps of 4
        // A frag: pixels k0,k0+1 of this lane's block (same image row)
        const int k0 = t * 4 + khalf;
        const float* ap = arow + (k0 >> 3) * 256 + (k0 & 7);
        v2f a;
        a.x = ap[0];
        a.y = ap[1];

        // B frags: one contiguous float2 per N-tile (ds_load_b64, imm offset)
        const v2f b0 = bp[(t * 8 + 0) * 16];
        const v2f b1 = bp[(t * 8 + 1) * 16];
        const v2f b2 = bp[(t * 8 + 2) * 16];
        const v2f b3 = bp[(t * 8 + 3) * 16];

        acc0 = __builtin_amdgcn_wmma_f32_16x16x4_f32(false, a, false, b0,
                                                     (short)0, acc0, false, false);
        acc1 = __builtin_amdgcn_wmma_f32_16x16x4_f32(false, a, false, b1,
                                                     (short)0, acc1, false, false);
        acc2 = __builtin_amdgcn_wmma_f32_16x16x4_f32(false, a, false, b2,
                                                     (short)0, acc2, false, false);
        acc3 = __builtin_amdgcn_wmma_f32_16x16x4_f32(false, a, false, b3,
                                                     (short)0, acc3, false, false);
    }

    // D layout: VGPR g -> lanes 0-15: (M=g, N=lane); lanes 16-31: (M=g+8, N=lane-16)
    const int mrow = mtile * 16 + ((lane >> 4) << 3);
    float* op = coeff + (size_t)mrow * 64 + col;
    #pragma unroll
    for (int g = 0; g < 8; ++g) {
        op[(size_t)g * 64 +  0] = acc0[g];
        op[(size_t)g * 64 + 16] = acc1[g];
        op[(size_t)g * 64 + 32] = acc2[g];
        op[(size_t)g * 64 + 48] = acc3[g];
    }
}

// ---------------------------------------------------------------------------
// Kernel 2: fused soft-CDF + adjacent-difference.
// One workgroup per (batch, channel). 1024 coefficients staged in LDS; the
// 8 waves split the 121 thresholds, the 32 lanes split the 1024 values,
// wave32 __shfl_xor reduction, then diff+scale written straight to d_out.
// ---------------------------------------------------------------------------
__global__ void __launch_bounds__(256) hist_kernel(
    const float* __restrict__ coeff,  // [16384,64]
    float* __restrict__ out)          // [16,120,64]
{
    const int b = blockIdx.x >> 6;    // batch
    const int k = blockIdx.x & 63;    // channel

    __shared__ float sx[1024];
    __shared__ float scdf[128];

    const int tid = threadIdx.x;
    #pragma unroll
    for (int i = 0; i < 4; ++i) {
        const int s = tid + i * 256;
        sx[s] = coeff[(size_t)(b * 1024 + s) * 64 + k];
    }
    __syncthreads();

    const int wave = tid >> 5;
    const int lane = tid & 31;
    const float gamma = 1000000.0f;

    for (int bin = wave; bin < 121; bin += 8) {
        const float thr = (float)(bin - 60);
        float acc = 0.0f;
        #pragma unroll
        for (int i = 0; i < 32; ++i) {
            const float x = sx[i * 32 + lane];
            const float z = gamma * (x - thr);
            acc += 1.0f / (1.0f + __expf(-z));   // sigmoid(gamma*(x-thr))
        }
        #pragma unroll
        for (int off = 16; off > 0; off >>= 1)
            acc += __shfl_xor(acc, off, 32);
        if (lane == 0) scdf[bin] = acc;
    }
    __syncthreads();

    if (tid < 120) {
        out[((size_t)b * 120 + tid) * 64 + k] =
            (scdf[tid] - scdf[tid + 1]) * (1.0f / 1024.0f);
    }
}

extern "C" void kernel_launch(void* const* d_in, const int* in_sizes, int n_in,
                              void* d_out, int out_size, void* d_ws, size_t ws_size,
                              hipStream_t stream) {
    const float* in    = (const float*)d_in[0];   // [16,256,256,1]
    const float* basis = (const float*)d_in[1];   // [8,8,1,64]
    float* out   = (float*)d_out;                 // [16,120,64,1]
    float* coeff = (float*)d_ws;                  // 16384*64 f32 = 4 MB scratch

    dct_wmma_kernel<<<128, 256, 0, stream>>>(in, basis, coeff);
    hist_kernel<<<1024, 256, 0, stream>>>(coeff, out);
}